// SVDHead_43662637531271
// MI455X (gfx1250) — compile-verified
//
#include <hip/hip_runtime.h>
#include <hip/hip_bf16.h>
#include <math.h>

typedef __attribute__((ext_vector_type(2))) float v2f;
typedef __attribute__((ext_vector_type(4))) float f4v;
typedef __attribute__((ext_vector_type(8))) float v8f;

#define NPTS 16384
#define BLK  256

// One block per batch. Streams src/tgt rows (coalesced float4 NT loads),
// accumulates 15 partial sums per thread, reduces across the block with
// chained V_WMMA_F32_16X16X4_F32 (B = ones => row-sum engine), then thread 0
// solves the 4x4 Horn quaternion eigenproblem and writes R (9) and t (3).
__global__ __launch_bounds__(256)
void svdhead_kernel(const float* __restrict__ src,
                    const float* __restrict__ tgt,
                    float* __restrict__ out, int B) {
  __shared__ float P[16][257];   // 15 partial quantities (+1 zero row) x 256 threads, padded
  __shared__ float Qs[16];       // final 15 block-wide sums

  const int b   = blockIdx.x;
  const int tid = threadIdx.x;
  const float* sb = src + (size_t)b * 3 * NPTS;
  const float* tb = tgt + (size_t)b * 3 * NPTS;

  float h00 = 0.f, h01 = 0.f, h02 = 0.f;
  float h10 = 0.f, h11 = 0.f, h12 = 0.f;
  float h20 = 0.f, h21 = 0.f, h22 = 0.f;
  float ss0 = 0.f, ss1 = 0.f, ss2 = 0.f;
  float ts0 = 0.f, ts1 = 0.f, ts2 = 0.f;

#pragma unroll 2
  for (int i = tid * 4; i < NPTS; i += BLK * 4) {
    f4v sx = __builtin_nontemporal_load((const f4v*)(sb + i));
    f4v sy = __builtin_nontemporal_load((const f4v*)(sb + NPTS + i));
    f4v sz = __builtin_nontemporal_load((const f4v*)(sb + 2 * NPTS + i));
    f4v tx = __builtin_nontemporal_load((const f4v*)(tb + i));
    f4v ty = __builtin_nontemporal_load((const f4v*)(tb + NPTS + i));
    f4v tz = __builtin_nontemporal_load((const f4v*)(tb + 2 * NPTS + i));
#pragma unroll
    for (int e = 0; e < 4; ++e) {
      float a0 = sx[e], a1 = sy[e], a2 = sz[e];
      float b0 = tx[e], b1 = ty[e], b2 = tz[e];
      ss0 += a0; ss1 += a1; ss2 += a2;
      ts0 += b0; ts1 += b1; ts2 += b2;
      h00 = fmaf(a0, b0, h00); h01 = fmaf(a0, b1, h01); h02 = fmaf(a0, b2, h02);
      h10 = fmaf(a1, b0, h10); h11 = fmaf(a1, b1, h11); h12 = fmaf(a1, b2, h12);
      h20 = fmaf(a2, b0, h20); h21 = fmaf(a2, b1, h21); h22 = fmaf(a2, b2, h22);
    }
  }

  P[0][tid] = h00;  P[1][tid] = h01;  P[2][tid] = h02;
  P[3][tid] = h10;  P[4][tid] = h11;  P[5][tid] = h12;
  P[6][tid] = h20;  P[7][tid] = h21;  P[8][tid] = h22;
  P[9][tid]  = ss0; P[10][tid] = ss1; P[11][tid] = ss2;
  P[12][tid] = ts0; P[13][tid] = ts1; P[14][tid] = ts2;
  P[15][tid] = 0.f;
  __syncthreads();

  // ---- WMMA block reduction: D += A_chunk(16x4) * ones(4x16) over 64 chunks.
  // B all-ones makes D[m][*] = total row-sum; A just needs each (lane,vgpr)
  // to hold a unique column of row (lane&15).
  if (tid < 32) {
    v8f acc = {0.f, 0.f, 0.f, 0.f, 0.f, 0.f, 0.f, 0.f};
    v2f bone; bone.x = 1.0f; bone.y = 1.0f;
    const int m  = tid & 15;
    const int ko = (tid >> 4) << 1;
#pragma unroll 1
    for (int c = 0; c < BLK; c += 4) {
      v2f a;
      a.x = P[m][c + ko];
      a.y = P[m][c + ko + 1];
      acc = __builtin_amdgcn_wmma_f32_16x16x4_f32(
          /*neg_a=*/false, a, /*neg_b=*/false, bone,
          /*c_mod=*/(short)0, acc, /*reuse_a=*/false, /*reuse_b=*/false);
    }
    // D layout: VGPR r -> M=r (lanes 0-15, N=lane) / M=r+8 (lanes 16-31).
    if (tid == 0) {
#pragma unroll
      for (int r = 0; r < 8; ++r) Qs[r] = acc[r];
    }
    if (tid == 16) {
#pragma unroll
      for (int r = 0; r < 8; ++r) Qs[8 + r] = acc[r];
    }
  }
  __syncthreads();

  // ---- per-batch solve (Horn quaternion == Kabsch with reflection fix)
  if (tid == 0) {
    const float invN = 1.0f / (float)NPTS;
    float su0 = Qs[9],  su1 = Qs[10], su2 = Qs[11];
    float tu0 = Qs[12], tu1 = Qs[13], tu2 = Qs[14];
    float Sxx = Qs[0] - su0 * tu0 * invN;
    float Sxy = Qs[1] - su0 * tu1 * invN;
    float Sxz = Qs[2] - su0 * tu2 * invN;
    float Syx = Qs[3] - su1 * tu0 * invN;
    float Syy = Qs[4] - su1 * tu1 * invN;
    float Syz = Qs[5] - su1 * tu2 * invN;
    float Szx = Qs[6] - su2 * tu0 * invN;
    float Szy = Qs[7] - su2 * tu1 * invN;
    float Szz = Qs[8] - su2 * tu2 * invN;

    float K4[4][4];
    K4[0][0] = Sxx + Syy + Szz;
    K4[0][1] = K4[1][0] = Syz - Szy;
    K4[0][2] = K4[2][0] = Szx - Sxz;
    K4[0][3] = K4[3][0] = Sxy - Syx;
    K4[1][1] = Sxx - Syy - Szz;
    K4[1][2] = K4[2][1] = Sxy + Syx;
    K4[1][3] = K4[3][1] = Szx + Sxz;
    K4[2][2] = -Sxx + Syy - Szz;
    K4[2][3] = K4[3][2] = Syz + Szy;
    K4[3][3] = -Sxx - Syy + Szz;

    float fr = 1e-30f;
#pragma unroll
    for (int i = 0; i < 4; ++i)
#pragma unroll
      for (int j = 0; j < 4; ++j) fr += K4[i][j] * K4[i][j];
    fr = sqrtf(fr);

    // Shifted, scaled: spectrum in [0,1]; repeated squaring -> projector qq^T.
    float Mp[4][4];
    const float sc = 1.0f / (2.0f * fr);
#pragma unroll
    for (int i = 0; i < 4; ++i)
#pragma unroll
      for (int j = 0; j < 4; ++j)
        Mp[i][j] = (K4[i][j] + (i == j ? fr : 0.0f)) * sc;

#pragma unroll 1
    for (int it = 0; it < 20; ++it) {
      float T[4][4];
#pragma unroll
      for (int i = 0; i < 4; ++i)
#pragma unroll
        for (int j = 0; j < 4; ++j) {
          float a = 0.f;
#pragma unroll
          for (int k = 0; k < 4; ++k) a = fmaf(Mp[i][k], Mp[k][j], a);
          T[i][j] = a;
        }
      float nf = 1e-37f;
#pragma unroll
      for (int i = 0; i < 4; ++i)
#pragma unroll
        for (int j = 0; j < 4; ++j) nf += T[i][j] * T[i][j];
      const float inv = 1.0f / sqrtf(nf);
#pragma unroll
      for (int i = 0; i < 4; ++i)
#pragma unroll
        for (int j = 0; j < 4; ++j) Mp[i][j] = T[i][j] * inv;
    }

    // Column with the largest diagonal entry of the projector ~ eigenvector.
    float q0 = Mp[0][0], q1 = Mp[1][0], q2 = Mp[2][0], q3 = Mp[3][0];
    float best = Mp[0][0];
    if (Mp[1][1] > best) { best = Mp[1][1]; q0 = Mp[0][1]; q1 = Mp[1][1]; q2 = Mp[2][1]; q3 = Mp[3][1]; }
    if (Mp[2][2] > best) { best = Mp[2][2]; q0 = Mp[0][2]; q1 = Mp[1][2]; q2 = Mp[2][2]; q3 = Mp[3][2]; }
    if (Mp[3][3] > best) { best = Mp[3][3]; q0 = Mp[0][3]; q1 = Mp[1][3]; q2 = Mp[2][3]; q3 = Mp[3][3]; }

    // Two polish power iterations on (K4 + fr*I).
#pragma unroll
    for (int it = 0; it < 2; ++it) {
      float r0 = fr * q0 + K4[0][0] * q0 + K4[0][1] * q1 + K4[0][2] * q2 + K4[0][3] * q3;
      float r1 = fr * q1 + K4[1][0] * q0 + K4[1][1] * q1 + K4[1][2] * q2 + K4[1][3] * q3;
      float r2 = fr * q2 + K4[2][0] * q0 + K4[2][1] * q1 + K4[2][2] * q2 + K4[2][3] * q3;
      float r3 = fr * q3 + K4[3][0] * q0 + K4[3][1] * q1 + K4[3][2] * q2 + K4[3][3] * q3;
      const float inv = 1.0f / sqrtf(fmaf(r0, r0, fmaf(r1, r1, fmaf(r2, r2, r3 * r3))) + 1e-37f);
      q0 = r0 * inv; q1 = r1 * inv; q2 = r2 * inv; q3 = r3 * inv;
    }

    const float w = q0, x = q1, y = q2, z = q3;
    const float R00 = 1.f - 2.f * (y * y + z * z), R01 = 2.f * (x * y - w * z), R02 = 2.f * (x * z + w * y);
    const float R10 = 2.f * (x * y + w * z), R11 = 1.f - 2.f * (x * x + z * z), R12 = 2.f * (y * z - w * x);
    const float R20 = 2.f * (x * z - w * y), R21 = 2.f * (y * z + w * x), R22 = 1.f - 2.f * (x * x + y * y);

    float* Ro = out + (size_t)b * 9;
    Ro[0] = R00; Ro[1] = R01; Ro[2] = R02;
    Ro[3] = R10; Ro[4] = R11; Ro[5] = R12;
    Ro[6] = R20; Ro[7] = R21; Ro[8] = R22;

    const float ms0 = su0 * invN, ms1 = su1 * invN, ms2 = su2 * invN;
    const float mt0 = tu0 * invN, mt1 = tu1 * invN, mt2 = tu2 * invN;
    float* To = out + (size_t)B * 9 + (size_t)b * 3;
    To[0] = mt0 - (R00 * ms0 + R01 * ms1 + R02 * ms2);
    To[1] = mt1 - (R10 * ms0 + R11 * ms1 + R12 * ms2);
    To[2] = mt2 - (R20 * ms0 + R21 * ms1 + R22 * ms2);
  }
}

extern "C" void kernel_launch(void* const* d_in, const int* in_sizes, int n_in,
                              void* d_out, int out_size, void* d_ws, size_t ws_size,
                              hipStream_t stream) {
  const float* src = (const float*)d_in[0];
  const float* tgt = (const float*)d_in[1];
  float* out = (float*)d_out;
  const int B = in_sizes[0] / (3 * NPTS);   // 1024
  svdhead_kernel<<<dim3(B), dim3(BLK), 0, stream>>>(src, tgt, out, B);
}